// Mixup_21689584845010
// MI455X (gfx1250) — compile-verified
//
#include <hip/hip_runtime.h>
#include <stdint.h>

// Clang native vector type: required for __builtin_nontemporal_* and lowers
// to b128 vector memory ops.
typedef float v4f __attribute__((ext_vector_type(4)));

// Problem constants (match reference)
constexpr int kB    = 32;
constexpr int kT    = 2048;
constexpr int kF    = 512;      // 128 float4 per row
constexpr int kM    = 4;        // MAX_NUM_MIX
constexpr int TROWS = 4;        // t-rows per block

// ---------------------------------------------------------------------------
// Kernel 1: per-batch scale normalization (B*M = 128 floats of work).
// scale[b,j] = (j < n) ? sc[b,j]*lam[b,j] / sum_{i<n} sc[b,i] : 0
// with lam = 0.1 + 0.3*lu, sc = 0.001 + 0.999*su, n = num_mixup_raw[b]+1
// ---------------------------------------------------------------------------
__global__ __launch_bounds__(32)
void mixup_scales_kernel(const int*   __restrict__ num_mixup_raw,
                         const float* __restrict__ lambda_u,
                         const float* __restrict__ scales_u,
                         float*       __restrict__ scales_out)
{
    const int b = threadIdx.x;
    if (b >= kB) return;
    const int n = num_mixup_raw[b] + 1;   // [1, kM]

    float lam[kM], sc[kM];
    float denom = 0.0f;
#pragma unroll
    for (int j = 0; j < kM; ++j) {
        lam[j] = 0.1f   + 0.3f   * lambda_u[b * kM + j];
        sc[j]  = 0.001f + 0.999f * scales_u[b * kM + j];
        if (j < n) denom += sc[j];
    }
    const float inv = 1.0f / denom;
#pragma unroll
    for (int j = 0; j < kM; ++j) {
        scales_out[b * kM + j] = (j < n) ? sc[j] * lam[j] * inv : 0.0f;
    }
}

// ---------------------------------------------------------------------------
// Kernel 2: streaming mixup.
// grid = (kT/TROWS, kB), block = 128 threads (4 wave32), thread -> one float4.
// out[b,t,:] = input[b,t,:] + mask[b,t] * sum_j scale[b,j]*cache[start[b,j]+t,:]
// input/out are single-use streams -> non-temporal b128; cache rows stay RT
// so overlapping gather windows can hit the 192MB L2.
// ---------------------------------------------------------------------------
__global__ __launch_bounds__(128)
void mixup_main_kernel(const float*         __restrict__ input,
                       const unsigned char* __restrict__ seq_mask,   // bool, 1B/elem
                       const float*         __restrict__ cache,
                       const int*           __restrict__ starts,     // [B, M]
                       const float*         __restrict__ scales,     // [B, M] (ws)
                       float*               __restrict__ out)
{
    const int b  = blockIdx.y;              // uniform -> SGPR
    const int t0 = blockIdx.x * TROWS;      // uniform -> SGPR
    const int fi = threadIdx.x;             // float4 index within row

    // Uniform per-block parameters: compiler scalarizes these (s_load).
    const float s0 = scales[b * kM + 0];
    const float s1 = scales[b * kM + 1];
    const float s2 = scales[b * kM + 2];
    const float s3 = scales[b * kM + 3];
    const int   r0 = starts[b * kM + 0];
    const int   r1 = starts[b * kM + 1];
    const int   r2 = starts[b * kM + 2];
    const int   r3 = starts[b * kM + 3];

#pragma unroll
    for (int dt = 0; dt < TROWS; ++dt) {
        const int t = t0 + dt;
        const float mf = seq_mask[(size_t)b * kT + t] ? 1.0f : 0.0f;

        const v4f* pin = (const v4f*)(input + (size_t)(b * kT + t) * kF) + fi;
        const v4f* c0  = (const v4f*)(cache + (size_t)(r0 + t) * kF) + fi;
        const v4f* c1  = (const v4f*)(cache + (size_t)(r1 + t) * kF) + fi;
        const v4f* c2  = (const v4f*)(cache + (size_t)(r2 + t) * kF) + fi;
        const v4f* c3  = (const v4f*)(cache + (size_t)(r3 + t) * kF) + fi;

        // Single-use stream: non-temporal load (TH=NT); cache[] stays regular.
        const v4f vi = __builtin_nontemporal_load(pin);
        const v4f v0 = *c0;
        const v4f v1 = *c1;
        const v4f v2 = *c2;
        const v4f v3 = *c3;

        v4f acc;
        acc.x = fmaf(s3, v3.x, fmaf(s2, v2.x, fmaf(s1, v1.x, s0 * v0.x)));
        acc.y = fmaf(s3, v3.y, fmaf(s2, v2.y, fmaf(s1, v1.y, s0 * v0.y)));
        acc.z = fmaf(s3, v3.z, fmaf(s2, v2.z, fmaf(s1, v1.z, s0 * v0.z)));
        acc.w = fmaf(s3, v3.w, fmaf(s2, v2.w, fmaf(s1, v1.w, s0 * v0.w)));

        v4f vo;
        vo.x = fmaf(mf, acc.x, vi.x);
        vo.y = fmaf(mf, acc.y, vi.y);
        vo.z = fmaf(mf, acc.z, vi.z);
        vo.w = fmaf(mf, acc.w, vi.w);

        v4f* pout = (v4f*)(out + (size_t)(b * kT + t) * kF) + fi;
        __builtin_nontemporal_store(vo, pout);   // NT store: write-once stream
    }
}

// ---------------------------------------------------------------------------
// Entry point
// Inputs (setup_inputs order):
//   0 input[B,T,F] f32 | 1 sequence_mask[B,T] bool | 2 cache[BUF,F] f32
//   3 start_indices[B,M] i32 | 4 num_mixup_raw[B] i32
//   5 lambda_u[B,M] f32 | 6 scales_u[B,M] f32
// Output: [B,T,F] f32
// ---------------------------------------------------------------------------
extern "C" void kernel_launch(void* const* d_in, const int* in_sizes, int n_in,
                              void* d_out, int out_size, void* d_ws, size_t ws_size,
                              hipStream_t stream)
{
    const float*         input  = (const float*)d_in[0];
    const unsigned char* mask   = (const unsigned char*)d_in[1];
    const float*         cache  = (const float*)d_in[2];
    const int*           starts = (const int*)d_in[3];
    const int*           nmix   = (const int*)d_in[4];
    const float*         lu     = (const float*)d_in[5];
    const float*         su     = (const float*)d_in[6];
    float*               out    = (float*)d_out;
    float*               scales = (float*)d_ws;   // B*M floats = 512 B scratch

    mixup_scales_kernel<<<dim3(1), dim3(32), 0, stream>>>(nmix, lu, su, scales);

    dim3 grid(kT / TROWS, kB);   // (512, 32) = 16384 blocks
    dim3 block(128);             // 4 wave32 per block; thread = one float4 of a row
    mixup_main_kernel<<<grid, block, 0, stream>>>(input, mask, cache, starts,
                                                  scales, out);
}